// PlanLayer_66314295050738
// MI455X (gfx1250) — compile-verified
//
#include <hip/hip_runtime.h>
#include <cstdint>

typedef float v2f __attribute__((ext_vector_type(2)));
typedef float v8f __attribute__((ext_vector_type(8)));

#define B_   16
#define T_   128
#define D_   14
#define N_   (T_*D_)            // 1792
#define MGP  ((T_-1)*D_)        // 1778
#define MOB  (T_*2)             // 256
#define M_   (MGP+MOB)          // 2034
#define NNZ_A 53368
#define NNZ_K 25404
#define HGP_STRIDE (D_*2*D_)    // 392 floats per GP factor Jacobian (14x28 row-major)
#define HOB_OFF    (MGP*2*D_)   // 49784
#define KGP_STRIDE (D_*D_)      // 196
#define KOB_OFF    ((T_-1)*D_*D_) // 24892
#define DELTA 0.1f

// workspace layout in floats, per batch (16x16 padded blocks)
#define GLL 0
#define GLR (GLL + (T_-1)*256)
#define GRL (GLR + (T_-1)*256)
#define GRR (GRL + (T_-1)*256)
#define OBO (GRR + (T_-1)*256)
#define RGP (OBO + T_*256)       // 28 rhs values per GP factor, stride 32
#define ROB (RGP + (T_-1)*32)    // 16 per state
#define WO  (ROB + T_*16)        // Thomas W_t blocks
#define YO  (WO  + T_*256)      // Thomas y_t vectors
#define WSB (YO  + T_*16)       // per-batch stride (203744 floats)

// Async copy of 16 bytes/lane from global to LDS (GVS mode: SGPR64 base +
// 32-bit VGPR byte offset; VDST VGPR = LDS byte address). Tracked by ASYNCcnt.
#define ASYNC_B128(ldsPtr, gPtr, idx)                                          \
    do {                                                                       \
        unsigned _la = (unsigned)(uintptr_t)(ldsPtr) + (unsigned)(idx) * 16u;  \
        unsigned _vo = (unsigned)(idx) * 16u;                                  \
        asm volatile("global_load_async_to_lds_b128 %0, %1, %2"                \
                     :: "v"(_la), "v"(_vo), "s"(gPtr) : "memory");             \
    } while (0)

#define WAIT_ASYNC() asm volatile("s_wait_asynccnt 0x0" ::: "memory")

// ---------------------------------------------------------------------------
// Kernel 1: per GP factor, G = Hpad^T * Kpad * Hpad via f32 WMMA (16x16x4),
// plus rhs contribution r = H^T (K b_f).  One wave per (batch, factor).
// ---------------------------------------------------------------------------
__global__ __launch_bounds__(256) void gp_factor_kernel(
    const float* __restrict__ Hv, const float* __restrict__ Kv,
    const float* __restrict__ bv, float* __restrict__ ws)
{
    __shared__ float shH[8 * 512];   // 16x32 per wave
    __shared__ float shK[8 * 256];   // 16x16 per wave
    __shared__ float shP[8 * 512];   // 16x32 per wave
    __shared__ float shV[8 * 16];
    __shared__ float shB[8 * 16];

    const int w    = threadIdx.x >> 5;
    const int lane = threadIdx.x & 31;
    const int inst = blockIdx.x * 8 + w;     // 254 blocks * 8 waves == 16*127 exactly
    const int bb   = inst / (T_ - 1);
    const int f    = inst % (T_ - 1);

    float* sH = shH + w * 512;
    float* sK = shK + w * 256;
    float* sP = shP + w * 512;
    float* sV = shV + w * 16;
    float* sB = shB + w * 16;

    const float* Hsrc = Hv + bb * NNZ_A + f * HGP_STRIDE;
    const float* Ksrc = Kv + bb * NNZ_K + f * KGP_STRIDE;
    if (f + 1 < T_ - 1) __builtin_prefetch(Hsrc + HGP_STRIDE, 0, 0);  // global_prefetch_b8

    for (int idx = lane; idx < 512; idx += 32) {
        int r = idx >> 5, c = idx & 31;
        sH[idx] = (r < D_ && c < 2 * D_) ? Hsrc[r * (2 * D_) + c] : 0.f;
    }
    for (int idx = lane; idx < 256; idx += 32) {
        int r = idx >> 4, c = idx & 15;
        sK[idx] = (r < D_ && c < D_) ? Ksrc[r * D_ + c] : 0.f;
    }
    if (lane < 16) sB[lane] = (lane < D_) ? bv[bb * M_ + f * D_ + lane] : 0.f;
    __syncthreads();

    // rhs: v = K b_f ; r = H^T v  (28 values -> states f and f+1)
    if (lane < 16) {
        float s = 0.f;
        for (int k = 0; k < 16; k++) s += sK[lane * 16 + k] * sB[k];
        sV[lane] = s;
    }
    __syncthreads();
    float* wsb = ws + (size_t)bb * WSB;
    if (lane < 2 * D_) {
        float s = 0.f;
        for (int i = 0; i < 16; i++) s += sH[i * 32 + lane] * sV[i];
        wsb[RGP + f * 32 + lane] = s;
    }

    // WMMA fragment coordinates (ISA 7.12.2, wave32, 32-bit operands)
    const int m  = lane & 15;      // M (A) / N (B,C,D) index
    const int h  = lane >> 4;      // lane half
    const int kb = h * 2;          // A-frag K base: lanes16-31 hold K=2,3

    // P = Kpad * Hpad : two 16x16 tiles, each 4 chained 16x16x4 WMMAs
    #pragma unroll
    for (int x = 0; x < 2; x++) {
        v8f acc = {};
        #pragma unroll
        for (int k0 = 0; k0 < 16; k0 += 4) {
            v2f a;   a[0]   = sK[m * 16 + k0 + kb];
                     a[1]   = sK[m * 16 + k0 + kb + 1];
            v2f bfr; bfr[0] = sH[(k0 + 2 * h)     * 32 + 16 * x + m];
                     bfr[1] = sH[(k0 + 2 * h + 1) * 32 + 16 * x + m];
            acc = __builtin_amdgcn_wmma_f32_16x16x4_f32(false, a, false, bfr,
                                                        (short)0, acc, false, false);
        }
        #pragma unroll
        for (int v = 0; v < 8; v++) sP[(v + 8 * h) * 32 + 16 * x + m] = acc[v];
    }
    __syncthreads();

    // G = Hpad^T * P : four 16x16 tiles -> Gll / Glr / Grl / Grr
    #pragma unroll
    for (int x = 0; x < 2; x++) {
        #pragma unroll
        for (int y = 0; y < 2; y++) {
            v8f acc = {};
            #pragma unroll
            for (int k0 = 0; k0 < 16; k0 += 4) {
                v2f a;   a[0]   = sH[(k0 + kb)     * 32 + 16 * x + m];  // H^T
                         a[1]   = sH[(k0 + kb + 1) * 32 + 16 * x + m];
                v2f bfr; bfr[0] = sP[(k0 + 2 * h)     * 32 + 16 * y + m];
                         bfr[1] = sP[(k0 + 2 * h + 1) * 32 + 16 * y + m];
                acc = __builtin_amdgcn_wmma_f32_16x16x4_f32(false, a, false, bfr,
                                                            (short)0, acc, false, false);
            }
            const int off = (x == 0) ? (y == 0 ? GLL : GLR) : (y == 0 ? GRL : GRR);
            float* dst = wsb + off + f * 256;
            #pragma unroll
            for (int v = 0; v < 8; v++) dst[(v + 8 * h) * 16 + m] = acc[v];
        }
    }
}

// ---------------------------------------------------------------------------
// Kernel 2: per state, OB = J^T K2 J (14x14, padded to 16x16) + rhs J^T K2 b2
// ---------------------------------------------------------------------------
__global__ __launch_bounds__(256) void obstacle_kernel(
    const float* __restrict__ Hv, const float* __restrict__ Kv,
    const float* __restrict__ bv, float* __restrict__ ws)
{
    const int bb  = blockIdx.x >> 7;
    const int s   = blockIdx.x & 127;
    const int tid = threadIdx.x;
    __shared__ float sJ[2 * D_], sK2[4], sb2[2];
    if (tid < 2 * D_) sJ[tid]  = Hv[bb * NNZ_A + HOB_OFF + s * 2 * D_ + tid];
    if (tid < 4)      sK2[tid] = Kv[bb * NNZ_K + KOB_OFF + s * 4 + tid];
    if (tid < 2)      sb2[tid] = bv[bb * M_ + MGP + 2 * s + tid];
    __syncthreads();

    const int i = tid >> 4, j = tid & 15;
    float val = 0.f;
    if (i < D_ && j < D_) {
        #pragma unroll
        for (int r = 0; r < 2; r++)
            #pragma unroll
            for (int c = 0; c < 2; c++)
                val += sJ[r * D_ + i] * sK2[r * 2 + c] * sJ[c * D_ + j];
    }
    float* wsb = ws + (size_t)bb * WSB;
    wsb[OBO + s * 256 + tid] = val;
    if (tid < 16) {
        float kb0 = sK2[0] * sb2[0] + sK2[1] * sb2[1];
        float kb1 = sK2[2] * sb2[0] + sK2[3] * sb2[1];
        float rv  = (tid < D_) ? sJ[tid] * kb0 + sJ[D_ + tid] * kb1 : 0.f;
        wsb[ROB + s * 16 + tid] = rv;
    }
}

// ---------------------------------------------------------------------------
// Kernel 3: block-Thomas solve, one workgroup per batch.
//   C_t = D_t - L_{t-1} W_{t-1};  W_t = C^-1 U_t;  y_t = C^-1 (r_t - L y_{t-1})
//   x_t = y_t - W_t x_{t+1}
// Block staging via global_load_async_to_lds_b128 (ASYNCcnt); Gauss-Jordan
// with ping-pong buffers -> exactly 1 barrier per pivot step.
// ---------------------------------------------------------------------------
__global__ __launch_bounds__(256) void solve_kernel(float* __restrict__ ws,
                                                    float* __restrict__ out)
{
    const int bb  = blockIdx.x;
    const int tid = threadIdx.x;
    float* wsb = ws + (size_t)bb * WSB;

    __shared__ float sAug0[16 * 32], sAug1[16 * 32];
    __shared__ float sL[256], sU[256], sOB[256], sGll[256], sGrr[256], sW[256];
    __shared__ float sy[16], sr[16], sxA[16], sxB[16];

    const int g64 = tid >> 6;      // async staging group
    const int l64 = tid & 63;
    const int r0  = tid >> 5;      // rows 0..7   (entry 1)
    const int c0  = tid & 31;      // augmented column 0..31
    const int r1  = r0 + 8;        // rows 8..15  (entry 2)
    const int i16 = tid >> 4, j16 = tid & 15;

    for (int t = 0; t < T_; t++) {
        // ---- phase 1: stage this step's blocks into LDS asynchronously ----
        if (g64 == 0 && t > 0) {
            ASYNC_B128(sL,   wsb + GRL + (t - 1) * 256, l64);
            ASYNC_B128(sGrr, wsb + GRR + (t - 1) * 256, l64);
        }
        if (g64 == 1 && t < T_ - 1) ASYNC_B128(sU,   wsb + GLR + t * 256, l64);
        if (g64 == 2)               ASYNC_B128(sOB,  wsb + OBO + t * 256, l64);
        if (g64 == 3 && t < T_ - 1) ASYNC_B128(sGll, wsb + GLL + t * 256, l64);
        if (t == 0)      { sL[tid] = 0.f; sGrr[tid] = 0.f; }
        if (t == T_ - 1) { sU[tid] = 0.f; sGll[tid] = 0.f; }
        WAIT_ASYNC();
        __syncthreads();                                      // B1

        // ---- phase 2: assemble augmented [C | I] into sAug0, rhs into sr ---
        {
            #pragma unroll
            for (int e = 0; e < 2; e++) {
                const int r = e ? r1 : r0;
                float v;
                if (c0 < 16) {
                    v = sOB[r * 16 + c0] + sGll[r * 16 + c0] + sGrr[r * 16 + c0];
                    if (r == c0) v += (r < D_) ? DELTA : 1.0f;
                    if (t > 0) {
                        float s = 0.f;
                        for (int k = 0; k < 16; k++) s += sL[r * 16 + k] * sW[k * 16 + c0];
                        v -= s;
                    }
                } else {
                    v = (r == c0 - 16) ? 1.f : 0.f;
                }
                sAug0[r * 32 + c0] = v;
            }
            if (tid < 16) {
                float r = wsb[ROB + t * 16 + tid];
                if (tid < D_) {
                    if (t < T_ - 1) r += wsb[RGP + t * 32 + tid];
                    if (t > 0)      r += wsb[RGP + (t - 1) * 32 + D_ + tid];
                }
                if (t > 0) {
                    float s = 0.f;
                    for (int k = 0; k < 16; k++) s += sL[tid * 16 + k] * sy[k];
                    r -= s;
                }
                sr[tid] = r;
            }
        }
        __syncthreads();                                      // B2

        // ---- phase 3: Gauss-Jordan, ping-pong (1 barrier / pivot step) -----
        #pragma unroll
        for (int k = 0; k < 16; k++) {
            const float* rd = (k & 1) ? sAug1 : sAug0;
            float*       wr = (k & 1) ? sAug0 : sAug1;
            float piv  = rd[k * 32 + k];
            float prow = rd[k * 32 + c0];
            float own0 = rd[r0 * 32 + c0], fac0 = rd[r0 * 32 + k];
            float own1 = rd[r1 * 32 + c0], fac1 = rd[r1 * 32 + k];
            float sp = prow * (1.0f / piv);
            wr[r0 * 32 + c0] = (r0 == k) ? sp : own0 - fac0 * sp;
            wr[r1 * 32 + c0] = (r1 == k) ? sp : own1 - fac1 * sp;
            __syncthreads();                                  // B3..B18
        }
        // 16 steps -> result back in sAug0; Cinv = sAug0[:, 16:32]

        // ---- phase 4: W_t = Cinv*U, y_t = Cinv*r~, store ------------------
        float wv = 0.f;
        for (int k = 0; k < 16; k++) wv += sAug0[i16 * 32 + 16 + k] * sU[k * 16 + j16];
        float yv = 0.f;
        if (tid < 16)
            for (int k = 0; k < 16; k++) yv += sAug0[tid * 32 + 16 + k] * sr[k];
        sW[tid] = wv;
        wsb[WO + t * 256 + tid] = wv;
        if (tid < 16) { sy[tid] = yv; wsb[YO + t * 16 + tid] = yv; }
        __syncthreads();                                      // B19
    }

    // ---- backward substitution (ping-pong x buffers, async W staging) -----
    float* xcur = sxA;
    float* xnxt = sxB;
    if (tid < 16) xcur[tid] = sy[tid];                 // x_{T-1} = y_{T-1}
    __syncthreads();
    if (tid < D_) out[bb * N_ + (T_ - 1) * D_ + tid] = xcur[tid];

    for (int t = T_ - 2; t >= 0; t--) {
        if (g64 == 0) ASYNC_B128(sW, wsb + WO + t * 256, l64);
        WAIT_ASYNC();
        __syncthreads();
        if (tid < 16) {
            float xv = wsb[YO + t * 16 + tid];
            for (int k = 0; k < 16; k++) xv -= sW[tid * 16 + k] * xcur[k];
            xnxt[tid] = xv;
        }
        __syncthreads();
        if (tid < D_) out[bb * N_ + t * D_ + tid] = xnxt[tid];
        float* tmp = xcur; xcur = xnxt; xnxt = tmp;
    }
}

// ---------------------------------------------------------------------------
extern "C" void kernel_launch(void* const* d_in, const int* in_sizes, int n_in,
                              void* d_out, int out_size, void* d_ws, size_t ws_size,
                              hipStream_t stream)
{
    (void)in_sizes; (void)n_in; (void)out_size; (void)ws_size;
    const float* Hv = (const float*)d_in[0];
    const float* Kv = (const float*)d_in[1];
    const float* bv = (const float*)d_in[2];
    float* ws  = (float*)d_ws;
    float* out = (float*)d_out;

    gp_factor_kernel<<<(B_ * (T_ - 1)) / 8, 256, 0, stream>>>(Hv, Kv, bv, ws); // 254 WGs
    obstacle_kernel<<<B_ * T_, 256, 0, stream>>>(Hv, Kv, bv, ws);              // 2048 WGs
    solve_kernel<<<B_, 256, 0, stream>>>(ws, out);                              // 16 WGs
}